// AttentionBlock_63273458205022
// MI455X (gfx1250) — compile-verified
//
#include <hip/hip_runtime.h>
#include <math.h>

// ---------------------------------------------------------------------------
// Attention block for MI455X (gfx1250), fp32 via V_WMMA_F32_16X16X4_F32,
// async-to-LDS staging (ASYNCcnt) with double buffering.
// GEMM waves now compute 32 rows x 128 cols (16 accumulator tiles) so each
// staged B fragment feeds two WMMAs (halves LDS traffic per matrix op).
// B=2, S=2048, E=2048, H=16, D=128.
// ---------------------------------------------------------------------------

typedef __attribute__((ext_vector_type(2))) float v2f;
typedef __attribute__((ext_vector_type(8))) float v8f;

#define EMB   2048
#define SEQ   2048
#define BATCH 2
#define NH    16
#define HD    128
#define MTOT  (BATCH * SEQ)      // 4096 rows
#define LPITCH 132               // padded LDS row pitch (floats): conflict-free

// D = A(16x4) * B(4x16) + C(16x16), fp32 everywhere.
__device__ __forceinline__ v8f wmma4(v2f a, v2f b, v8f c) {
    return __builtin_amdgcn_wmma_f32_16x16x4_f32(
        false, a, false, b, (short)0, c, false, false);
}

// Async global->LDS copy, 16B per lane, tracked by ASYNCcnt (gfx1250 path).
__device__ __forceinline__ void async_ld_b128(unsigned lds_byte_off, const void* g) {
    asm volatile("global_load_async_to_lds_b128 %0, %1, off"
                 :: "v"(lds_byte_off),
                    "v"((unsigned long long)(uintptr_t)g)
                 : "memory");
}
#define S_WAIT_ASYNC(n) asm volatile("s_wait_asynccnt " #n ::: "memory")

// LDS byte offset of a __shared__ element (generic ptr low 32 bits).
__device__ __forceinline__ unsigned lds_off(const void* p) {
    return (unsigned)(uintptr_t)p;
}

// Stage a 16x128 fp32 tile (row stride gstride floats in global) into LDS
// with padded pitch LPITCH. 128 threads x 4 async b128 = 8KB, coalesced.
__device__ __forceinline__ void stage_tile(float* sh, const float* g,
                                           int gstride, int tid) {
#pragma unroll
    for (int i = 0; i < 4; ++i) {
        const int off = i * 512 + tid * 4;
        const int r = off >> 7;
        const int c = off & 127;
        async_ld_b128(lds_off(&sh[r * LPITCH + c]), g + (size_t)r * gstride + c);
    }
}

// ---------------------------------------------------------------------------
// Kernel 1: Y = x @ W for W in {wq,wk,wv}; RoPE in-register for q,k.
// Block = 4 waves; each wave computes 32 rows x 128 cols (one head's cols),
// W tile staged once per block per 16-row K-chunk (async double-buffered).
// Output layout: [B, H, S, D]. Grid = (MTOT/128, NH, 3).
// ---------------------------------------------------------------------------
__global__ __launch_bounds__(128)
void qkv_gemm_rope(const float* __restrict__ x,
                   const float* __restrict__ wq,
                   const float* __restrict__ wk,
                   const float* __restrict__ wv,
                   float* __restrict__ qb,
                   float* __restrict__ kb,
                   float* __restrict__ vb)
{
    __shared__ float wsh[2][16 * LPITCH];

    const int tid  = threadIdx.x;
    const int lane = tid & 31;
    const int wave = tid >> 5;
    const int l16  = lane & 15;
    const int hi   = lane >> 4;          // A/B fragment half: K 0..1 vs 2..3
    const int z    = blockIdx.z;         // 0=q, 1=k, 2=v

    const float* W   = (z == 0) ? wq : (z == 1) ? wk : wv;
    float*       out = (z == 0) ? qb : (z == 1) ? kb : vb;

    const int rbase = blockIdx.x * 128 + wave * 32;  // global row (b*S + s)
    const int head  = blockIdx.y;
    const int cbase = head * HD;

    v8f acc[2][8] = {};                              // 2 row-halves x 8 tiles

    const float* Arow0 = x + (size_t)(rbase + l16) * EMB;
    const float* Arow1 = Arow0 + (size_t)16 * EMB;

    stage_tile(wsh[0], W + cbase, EMB, tid);
    for (int ch = 0; ch < 128; ++ch) {
        if (ch < 127) {
            stage_tile(wsh[(ch + 1) & 1],
                       W + (size_t)(ch + 1) * 16 * EMB + cbase, EMB, tid);
            S_WAIT_ASYNC(4);                         // chunk ch complete
        } else {
            S_WAIT_ASYNC(0);
        }
        __syncthreads();
        const float* Wt = wsh[ch & 1];
#pragma unroll
        for (int ks = 0; ks < 4; ++ks) {
            const int ka = ch * 16 + ks * 4 + 2 * hi;
            const v2f a0 = *(const v2f*)(Arow0 + ka);
            const v2f a1 = *(const v2f*)(Arow1 + ka);
            const int lr = ks * 4 + 2 * hi;
            const float* wr = Wt + lr * LPITCH + l16;
#pragma unroll
            for (int t = 0; t < 8; ++t) {
                v2f b;
                b.x = wr[t * 16];                    // W[ka  ][col]
                b.y = wr[t * 16 + LPITCH];           // W[ka+1][col]
                acc[0][t] = wmma4(a0, b, acc[0][t]); // B frag reused twice
                acc[1][t] = wmma4(a1, b, acc[1][t]);
            }
        }
        __syncthreads();
    }

    const int bidx = rbase >> 11;
    const int srow0 = rbase & (SEQ - 1);
    float* obase = out + (((size_t)bidx * NH + head) * SEQ) * HD;

    if (z < 2) {
        // RoPE: inv_freq[dd] = 10000^(-2*dd/128) = exp(-ln(1e4)/64 * dd)
#pragma unroll
        for (int t = 0; t < 4; ++t) {
            const int   dd   = t * 16 + l16;
            const float freq = __expf(-0.14391156f * (float)dd);
#pragma unroll
            for (int h2 = 0; h2 < 2; ++h2) {
#pragma unroll
                for (int v = 0; v < 8; ++v) {
                    const int   s   = srow0 + h2 * 16 + v + 8 * hi;
                    const float ang = (float)s * freq;
                    float sn, cs;
                    __sincosf(ang, &sn, &cs);
                    const float x1 = acc[h2][t][v];
                    const float x2 = acc[h2][t + 4][v];
                    obase[(size_t)s * HD + dd]      = x1 * cs - x2 * sn;
                    obase[(size_t)s * HD + dd + 64] = x1 * sn + x2 * cs;
                }
            }
        }
    } else {
#pragma unroll
        for (int h2 = 0; h2 < 2; ++h2)
#pragma unroll
            for (int t = 0; t < 8; ++t)
#pragma unroll
                for (int v = 0; v < 8; ++v) {
                    const int s = srow0 + h2 * 16 + v + 8 * hi;
                    obase[(size_t)s * HD + t * 16 + l16] = acc[h2][t][v];
                }
    }
}

// ---------------------------------------------------------------------------
// Kernel 2: causal flash attention. One block = 64 q-rows (4 waves x 16) of
// one (b, h). All waves march the block's full causal j-range so K/V tiles
// are async-staged to LDS once per block and shared; tiles beyond a wave's
// own causal limit are fully masked (exp -> 0, corr = 1: a no-op).
// Grid = (SEQ/64, NH, BATCH).
// ---------------------------------------------------------------------------
__global__ __launch_bounds__(128)
void flash_attn(const float* __restrict__ qb,
                const float* __restrict__ kb,
                const float* __restrict__ vb,
                float* __restrict__ ao)
{
    __shared__ float ksh[2][16 * LPITCH];
    __shared__ float vsh[2][16 * LPITCH];
    __shared__ float psh[4][16 * 17];                // P tile, pitch 17

    const int tid  = threadIdx.x;
    const int lane = tid & 31;
    const int wave = tid >> 5;
    const int l16  = lane & 15;
    const int hi   = lane >> 4;

    const int qt    = blockIdx.x * 4 + wave;         // this wave's q-tile
    const int jmax  = blockIdx.x * 4 + 3;            // block-wide causal limit
    const int head  = blockIdx.y;
    const int bidx  = blockIdx.z;
    const int qbase = qt * 16;

    const size_t bh = ((size_t)bidx * NH + head) * SEQ;
    const float* Q = qb + bh * HD;
    const float* K = kb + bh * HD;
    const float* V = vb + bh * HD;

    // Preload Q A-fragments (32 K-steps of 4 over D=128).
    v2f qa[32];
    {
        const float* qrow = Q + (size_t)(qbase + l16) * HD + 2 * hi;
#pragma unroll
        for (int ks = 0; ks < 32; ++ks)
            qa[ks] = *(const v2f*)(qrow + ks * 4);
    }

    v8f acc[8] = {};
    float mrow[8], lrow[8];
#pragma unroll
    for (int v = 0; v < 8; ++v) { mrow[v] = -INFINITY; lrow[v] = 0.f; }

    const float scale = 0.08838834764831845f;        // 1/sqrt(128)
    float* pl = &psh[wave][0];

    stage_tile(ksh[0], K, HD, tid);
    stage_tile(vsh[0], V, HD, tid);

    for (int j = 0; j <= jmax; ++j) {
        if (j < jmax) {
            const size_t nb = (size_t)(j + 1) * 16 * HD;
            stage_tile(ksh[(j + 1) & 1], K + nb, HD, tid);
            stage_tile(vsh[(j + 1) & 1], V + nb, HD, tid);
            S_WAIT_ASYNC(8);                         // tiles for j complete
        } else {
            S_WAIT_ASYNC(0);
        }
        __syncthreads();

        const float* Kt = ksh[j & 1];
        const float* Vt = vsh[j & 1];
        const int kbase = j * 16;

        // S = Q @ K^T : B[kk][n] = K[kbase+n][dk] read from LDS
        v8f sc = {};
        const float* kr = Kt + l16 * LPITCH + 2 * hi;
#pragma unroll
        for (int ks = 0; ks < 32; ++ks) {
            const v2f bf = *(const v2f*)(kr + ks * 4);
            sc = wmma4(qa[ks], bf, sc);
        }

        // scale + causal mask (handles fully-masked tiles too)
#pragma unroll
        for (int v = 0; v < 8; ++v) {
            const int qr = qbase + v + 8 * hi;
            const int kc = kbase + l16;
            float s = sc[v] * scale;
            if (kc > qr) s = -INFINITY;
            sc[v] = s;
        }

        // online softmax (row reductions within 16-lane C-layout groups)
#pragma unroll
        for (int v = 0; v < 8; ++v) {
            float m = sc[v];
            m = fmaxf(m, __shfl_xor(m, 1, 32));
            m = fmaxf(m, __shfl_xor(m, 2, 32));
            m = fmaxf(m, __shfl_xor(m, 4, 32));
            m = fmaxf(m, __shfl_xor(m, 8, 32));
            const float mnew = fmaxf(mrow[v], m);
            const float corr = __expf(mrow[v] - mnew);
            mrow[v] = mnew;

            const float p = __expf(sc[v] - mnew);
            sc[v] = p;
            float r = p;
            r += __shfl_xor(r, 1, 32);
            r += __shfl_xor(r, 2, 32);
            r += __shfl_xor(r, 4, 32);
            r += __shfl_xor(r, 8, 32);
            lrow[v] = lrow[v] * corr + r;
#pragma unroll
            for (int t = 0; t < 8; ++t) acc[t][v] *= corr;
        }

        // P tile: C layout -> LDS -> A layout (wave-local, DS in-order)
#pragma unroll
        for (int v = 0; v < 8; ++v)
            pl[(v + 8 * hi) * 17 + l16] = sc[v];

        // O += P(16x16) @ V(16x128) from LDS: 4 k-steps x 8 d-tiles
#pragma unroll
        for (int ks = 0; ks < 4; ++ks) {
            v2f pa;
            pa.x = pl[l16 * 17 + ks * 4 + 2 * hi];
            pa.y = pl[l16 * 17 + ks * 4 + 2 * hi + 1];
            const float* vr = Vt + (ks * 4 + 2 * hi) * LPITCH + l16;
#pragma unroll
            for (int t = 0; t < 8; ++t) {
                v2f bv;
                bv.x = vr[t * 16];
                bv.y = vr[t * 16 + LPITCH];
                acc[t] = wmma4(pa, bv, acc[t]);
            }
        }
        __syncthreads();
    }

    // normalize and store row-major [MTOT, EMB]
#pragma unroll
    for (int v = 0; v < 8; ++v) {
        const float inv  = 1.f / lrow[v];
        const int   srow = qbase + v + 8 * hi;
        const size_t ro  = ((size_t)(bidx * SEQ + srow)) * EMB + head * HD;
#pragma unroll
        for (int t = 0; t < 8; ++t)
            ao[ro + t * 16 + l16] = acc[t][v] * inv;
    }
}

// ---------------------------------------------------------------------------
// Kernel 3: out = attn_out @ wo, row-major output. Same 32-row waves and
// staging as kernel 1. Grid = (MTOT/128, NH, 1).
// ---------------------------------------------------------------------------
__global__ __launch_bounds__(128)
void out_gemm(const float* __restrict__ A,
              const float* __restrict__ W,
              float* __restrict__ out)
{
    __shared__ float wsh[2][16 * LPITCH];

    const int tid  = threadIdx.x;
    const int lane = tid & 31;
    const int wave = tid >> 5;
    const int l16  = lane & 15;
    const int hi   = lane >> 4;

    const int rbase = blockIdx.x * 128 + wave * 32;
    const int cbase = blockIdx.y * HD;

    v8f acc[2][8] = {};
    const float* Arow0 = A + (size_t)(rbase + l16) * EMB;
    const float* Arow1 = Arow0 + (size_t)16 * EMB;

    stage_tile(wsh[0], W + cbase, EMB, tid);
    for (int ch = 0; ch < 128; ++ch) {
        if (ch < 127) {
            stage_tile(wsh[(ch + 1) & 1],
                       W + (size_t)(ch + 1) * 16 * EMB + cbase, EMB, tid);
            S_WAIT_ASYNC(4);
        } else {
            S_WAIT_ASYNC(0);
        }
        __syncthreads();
        const float* Wt = wsh[ch & 1];
#pragma unroll
        for (int ks = 0; ks < 4; ++ks) {
            const int ka = ch * 16 + ks * 4 + 2 * hi;
            const v2f a0 = *(const v2f*)(Arow0 + ka);
            const v2f a1 = *(const v2f*)(Arow1 + ka);
            const int lr = ks * 4 + 2 * hi;
            const float* wr = Wt + lr * LPITCH + l16;
#pragma unroll
            for (int t = 0; t < 8; ++t) {
                v2f b;
                b.x = wr[t * 16];
                b.y = wr[t * 16 + LPITCH];
                acc[0][t] = wmma4(a0, b, acc[0][t]);
                acc[1][t] = wmma4(a1, b, acc[1][t]);
            }
        }
        __syncthreads();
    }

#pragma unroll
    for (int h2 = 0; h2 < 2; ++h2)
#pragma unroll
        for (int t = 0; t < 8; ++t)
#pragma unroll
            for (int v = 0; v < 8; ++v)
                out[(size_t)(rbase + h2 * 16 + v + 8 * hi) * EMB +
                    cbase + t * 16 + l16] = acc[h2][t][v];
}

// ---------------------------------------------------------------------------
extern "C" void kernel_launch(void* const* d_in, const int* in_sizes, int n_in,
                              void* d_out, int out_size, void* d_ws, size_t ws_size,
                              hipStream_t stream)
{
    const float* x  = (const float*)d_in[0];
    const float* wq = (const float*)d_in[1];
    const float* wk = (const float*)d_in[2];
    const float* wv = (const float*)d_in[3];
    const float* wo = (const float*)d_in[4];
    float*       out = (float*)d_out;

    const size_t mat_bytes = (size_t)MTOT * EMB * sizeof(float);   // 33.5 MB
    char* ws = (char*)d_ws;
    float* qb = (float*)(ws);                                      // [B,H,S,D]
    float* kb = (float*)(ws + mat_bytes);
    float* vb = (float*)(ws + 2 * mat_bytes);
    float* ab = (float*)(ws + 3 * mat_bytes);                      // [MTOT,EMB]

    dim3 gq(MTOT / 128, NH, 3);
    qkv_gemm_rope<<<gq, 128, 0, stream>>>(x, wq, wk, wv, qb, kb, vb);

    dim3 ga(SEQ / 64, NH, BATCH);
    flash_attn<<<ga, 128, 0, stream>>>(qb, kb, vb, ab);

    dim3 go(MTOT / 128, NH, 1);
    out_gemm<<<go, 128, 0, stream>>>(ab, wo, out);
}